// KVCache_45294725104260
// MI455X (gfx1250) — compile-verified
//
#include <hip/hip_runtime.h>

// KV-cache append (concat along seq axis), fp32.
// B=8, H=32, S_CACHE=4096, S_NEW=1, D=128. Pure streaming copy:
// ~2.15 GB HBM traffic, floor ~92 us @ 23.3 TB/s. Block-tiled so the
// tensor/bh select is scalar per block, and each wave issues 8 independent
// nontemporal b128 load/store pairs (NT keeps the stream out of the 192 MB L2).

typedef __attribute__((ext_vector_type(4))) float v4f;   // native vector, 16B

namespace {
constexpr unsigned kBH      = 8u * 32u;                  // 256
constexpr unsigned kRow4In  = 4096u * 128u / 4u;         // 131072 = 2^17
constexpr unsigned kRow4Out = 4097u * 128u / 4u;         // 131104
constexpr unsigned kCache4  = kBH * kRow4In;             // 33,554,432 = 2^25
constexpr unsigned kNew4    = kBH * (128u / 4u);         // 8,192 per tensor
constexpr unsigned kOutT4   = kBH * kRow4Out;            // output tensor stride

constexpr unsigned kThreads = 256;                       // 8 wave32s
constexpr unsigned kUnroll  = 8;
constexpr unsigned kTileV4  = kThreads * kUnroll;        // 2048 vec4s = 32 KB
constexpr unsigned kTilesPerTensor = kCache4 / kTileV4;  // 16384 (exact)
constexpr unsigned kBulkBlocks = 2u * kTilesPerTensor;   // K then V
constexpr unsigned kNewBlocks  = (2u * kNew4) / kTileV4; // 8 (exact)
} // namespace

__global__ __launch_bounds__(kThreads) void kv_concat_kernel(
    const v4f* __restrict__ key_cache,
    const v4f* __restrict__ value_cache,
    const v4f* __restrict__ key_new,
    const v4f* __restrict__ value_new,
    v4f* __restrict__ out)
{
    const unsigned blk = blockIdx.x;
    const unsigned t   = threadIdx.x;

    if (blk < kBulkBlocks) {
        // ---- Bulk cache copy: one contiguous 32 KB tile per block. ----
        const bool isV      = (blk >= kTilesPerTensor);
        const unsigned tile = blk - (isV ? kTilesPerTensor : 0u);
        const unsigned j0   = tile * kTileV4;     // vec4 index into cache
        const unsigned bh   = j0 >> 17;           // uniform across the tile
        // out row is 131104 = 2^17 + 2^5 vec4s -> out_idx = j + (bh << 5)
        const v4f* __restrict__ src = (isV ? value_cache : key_cache) + j0 + t;
        v4f* __restrict__ dst = out + (isV ? kOutT4 : 0u) + j0 + (bh << 5) + t;
#pragma unroll
        for (unsigned k = 0; k < kUnroll; ++k) {
            v4f v = __builtin_nontemporal_load(src + k * kThreads);
            __builtin_nontemporal_store(v, dst + k * kThreads);
        }
    } else {
        // ---- Append the new (s == S_CACHE) rows: 16,384 vec4s total. ----
        const unsigned base = (blk - kBulkBlocks) * kTileV4;
#pragma unroll
        for (unsigned k = 0; k < kUnroll; ++k) {
            const unsigned idx = base + t + k * kThreads;   // [0, 2*kNew4)
            const bool isV     = (idx >= kNew4);
            const unsigned j2  = idx - (isV ? kNew4 : 0u);
            const unsigned bh  = j2 >> 5;
            const unsigned d4  = j2 & 31u;
            v4f v = __builtin_nontemporal_load((isV ? value_new : key_new) + j2);
            __builtin_nontemporal_store(
                v, out + (isV ? kOutT4 : 0u) + bh * kRow4Out + kRow4In + d4);
        }
    }
}

extern "C" void kernel_launch(void* const* d_in, const int* in_sizes, int n_in,
                              void* d_out, int out_size, void* d_ws, size_t ws_size,
                              hipStream_t stream) {
    (void)in_sizes; (void)n_in; (void)out_size; (void)d_ws; (void)ws_size;

    const v4f* key_cache   = (const v4f*)d_in[0];
    const v4f* value_cache = (const v4f*)d_in[1];
    const v4f* key_new     = (const v4f*)d_in[2];
    const v4f* value_new   = (const v4f*)d_in[3];
    v4f* out = (v4f*)d_out;

    const unsigned blocks = kBulkBlocks + kNewBlocks;     // 32,776
    kv_concat_kernel<<<blocks, kThreads, 0, stream>>>(
        key_cache, value_cache, key_new, value_new, out);
}